// loss_v2_23252952941260
// MI455X (gfx1250) — compile-verified
//
#include <hip/hip_runtime.h>
#include <math.h>

// ---- problem constants (from reference) ----
#define NBATCH 32
#define NANCH  5
#define NCLS   20
#define NH     76
#define NW     76
#define NCELL  (NH * NW)          // 5776
#define MAXT   50
#define SIL_THRESH 0.6f

// phase-1 grid: 23 blocks of 256 threads per (b, a) pair; 160 (b,a) pairs
#define P1_BLK_PER_BA 23          // 23*256 = 5888 >= 5776
#define P1_NPART (P1_BLK_PER_BA * NBATCH * NANCH)   // 3680
#define NPART    (P1_NPART + NBATCH)                // 3712 = 58 * 64
#define FIN_ITERS (NPART / 64)                      // 58

typedef float v2f __attribute__((ext_vector_type(2)));
typedef float v8f __attribute__((ext_vector_type(8)));

__device__ __forceinline__ float sigf(float v) { return 1.0f / (1.0f + expf(-v)); }

__device__ __forceinline__ float iou_box(float ax, float ay, float aw, float ah,
                                         float bx, float by, float bw, float bh) {
    float uw = fmaxf(ax + aw * 0.5f, bx + bw * 0.5f) - fminf(ax - aw * 0.5f, bx - bw * 0.5f);
    float uh = fmaxf(ay + ah * 0.5f, by + bh * 0.5f) - fminf(ay - ah * 0.5f, by - bh * 0.5f);
    float cw = aw + bw - uw;
    float ch = ah + bh - uh;
    float carea = (cw <= 0.0f || ch <= 0.0f) ? 0.0f : cw * ch;
    return carea / (aw * ah + bw * bh - carea);
}

// ---------------------------------------------------------------------------
// Phase 1: dense no-object confidence loss over all 924,160 cells.
// Reads only channels 0..4 of each anchor slice (18.5 MB of 92 MB).
// One f32 partial per block -> ws[blockIdx.y * 23 + blockIdx.x].
// ---------------------------------------------------------------------------
__global__ void yolo_phase1(const float* __restrict__ output,
                            const float* __restrict__ target,
                            const float* __restrict__ anchors,
                            float* __restrict__ partials) {
    __shared__ float sgx[MAXT], sgy[MAXT], sgw[MAXT], sgh[MAXT];
    __shared__ int   sval[MAXT];
    __shared__ float red[256];

    const int ba = blockIdx.y;            // 0..159
    const int b  = ba / NANCH;
    const int a  = ba % NANCH;
    const int tid = threadIdx.x;

    if (tid < MAXT) {
        const float* tb = target + (size_t)b * (MAXT * 5) + tid * 5;
        float tx1 = tb[1];
        sgx[tid] = tx1   * (float)NW;
        sgy[tid] = tb[2] * (float)NH;
        sgw[tid] = tb[3] * (float)NW;
        sgh[tid] = tb[4] * (float)NH;
        sval[tid] = (tx1 != 0.0f);
    }
    __syncthreads();

    const float aw = anchors[2 * a];
    const float ah = anchors[2 * a + 1];

    const int s = blockIdx.x * 256 + tid; // cell within (b,a)
    float part = 0.0f;
    if (s < NCELL) {
        const int j = s / NW;
        const int i = s % NW;
        const float* base = output + ((size_t)(b * NANCH + a) * (5 + NCLS)) * NCELL + s;
        const float ox = base[0 * NCELL];
        const float oy = base[1 * NCELL];
        const float ow = base[2 * NCELL];
        const float oh = base[3 * NCELL];
        const float oc = base[4 * NCELL];

        const float px = sigf(ox) + (float)i;
        const float py = sigf(oy) + (float)j;
        const float pw = expf(ow) * aw;
        const float ph = expf(oh) * ah;
        const float conf = sigf(oc);

        float maxiou = 0.0f;
        for (int t = 0; t < MAXT; ++t) {
            if (sval[t]) {
                float v = iou_box(px, py, pw, ph, sgx[t], sgy[t], sgw[t], sgh[t]);
                maxiou = fmaxf(maxiou, v);
            }
        }
        part = (maxiou > SIL_THRESH) ? 0.0f : 0.5f * conf * conf; // NOOBJ_SCALE = 1
    }

    red[tid] = part;
    __syncthreads();
    for (int off = 128; off > 0; off >>= 1) {
        if (tid < off) red[tid] += red[tid + off];
        __syncthreads();
    }
    if (tid == 0) partials[blockIdx.y * P1_BLK_PER_BA + blockIdx.x] = red[0];
}

// ---------------------------------------------------------------------------
// Phase 2: per-GT-target sparse losses (coord, obj-conf, class CE) plus the
// correction subtracting the no-obj term phase 1 added at the selected cell.
// 32 blocks (one per image) x 64 threads; one partial per block.
// ---------------------------------------------------------------------------
__global__ void yolo_phase2(const float* __restrict__ output,
                            const float* __restrict__ target,
                            const float* __restrict__ anchors,
                            float* __restrict__ partials) {
    __shared__ float sgx[MAXT], sgy[MAXT], sgw[MAXT], sgh[MAXT];
    __shared__ int   sval[MAXT], scls[MAXT];
    __shared__ float red[64];

    const int b = blockIdx.x;
    const int t = threadIdx.x;

    if (t < MAXT) {
        const float* tb = target + (size_t)b * (MAXT * 5) + t * 5;
        float tx1 = tb[1];
        sgx[t] = tx1   * (float)NW;
        sgy[t] = tb[2] * (float)NH;
        sgw[t] = tb[3] * (float)NW;
        sgh[t] = tb[4] * (float)NH;
        sval[t] = (tx1 != 0.0f);
        int c = (int)tb[0];
        scls[t] = (c < 0) ? 0 : (c > NCLS - 1 ? NCLS - 1 : c);
    }
    __syncthreads();

    float part = 0.0f;
    if (t < MAXT && sval[t]) {
        const float gx = sgx[t], gy = sgy[t], gw = sgw[t], gh = sgh[t];

        // best anchor: argmax IoU of (0,0,gw,gh) vs (0,0,aw,ah) = min/min over union
        int best = 0;
        float bi = -1.0f;
        for (int a = 0; a < NANCH; ++a) {
            float aw_ = anchors[2 * a], ah_ = anchors[2 * a + 1];
            float inter = fminf(gw, aw_) * fminf(gh, ah_);
            float u = gw * gh + aw_ * ah_ - inter;
            float v = inter / u;
            if (v > bi) { bi = v; best = a; }
        }
        int gi = (int)gx; gi = gi < 0 ? 0 : (gi > NW - 1 ? NW - 1 : gi);
        int gj = (int)gy; gj = gj < 0 ? 0 : (gj > NH - 1 ? NH - 1 : gj);

        const float awb = anchors[2 * best], ahb = anchors[2 * best + 1];
        const float* base = output + ((size_t)(b * NANCH + best) * (5 + NCLS)) * NCELL
                                   + (size_t)gj * NW + gi;
        const float ox = base[0 * NCELL];
        const float oy = base[1 * NCELL];
        const float ow = base[2 * NCELL];
        const float oh = base[3 * NCELL];
        const float oc = base[4 * NCELL];

        const float x = sigf(ox), y = sigf(oy), conf = sigf(oc);
        const float txv = gx - (float)gi;
        const float tyv = gy - (float)gj;
        const float twv = logf(gw / awb);
        const float thv = logf(gh / ahb);

        // coord losses (COORD_SCALE = 1, coord_mask = 1)
        part += 0.5f * ((x - txv) * (x - txv) + (y - tyv) * (y - tyv) +
                        (ow - twv) * (ow - twv) + (oh - thv) * (oh - thv));

        // object confidence: conf_mask = 5 -> 2.5*(conf - tconf)^2
        const float px = x + (float)gi;
        const float py = y + (float)gj;
        const float pw = expf(ow) * awb;
        const float ph = expf(oh) * ahb;
        const float tconf = iou_box(gx, gy, gw, gh, px, py, pw, ph);
        part += 2.5f * (conf - tconf) * (conf - tconf);

        // subtract the no-obj term phase 1 added at this cell (same math/order)
        float maxiou = 0.0f;
        for (int u = 0; u < MAXT; ++u) {
            if (sval[u]) {
                float v = iou_box(px, py, pw, ph, sgx[u], sgy[u], sgw[u], sgh[u]);
                maxiou = fmaxf(maxiou, v);
            }
        }
        if (!(maxiou > SIL_THRESH)) part -= 0.5f * conf * conf;

        // class cross-entropy (CLASS_SCALE = 1, cls_mask = 1)
        const float* lb = base + (size_t)5 * NCELL;
        float m = lb[0];
        for (int c = 1; c < NCLS; ++c) m = fmaxf(m, lb[(size_t)c * NCELL]);
        float sum = 0.0f;
        for (int c = 0; c < NCLS; ++c) sum += expf(lb[(size_t)c * NCELL] - m);
        const float lse = logf(sum) + m;
        part += lse - lb[(size_t)scls[t] * NCELL];
    }

    red[t] = part;
    __syncthreads();
    for (int off = 32; off > 0; off >>= 1) {
        if (t < off) red[t] += red[t + off];
        __syncthreads();
    }
    if (t == 0) partials[P1_NPART + b] = red[0];
}

// ---------------------------------------------------------------------------
// Phase 3: single-wave32 deterministic reduction of 3712 partials using
// V_WMMA_F32_16X16X4_F32 with A = ones: D[m,n] = sum_k B[k,n] + C[m,n].
// Each WMMA folds 64 f32 partials (2 B-VGPRs x 32 lanes) into the f32
// accumulator; 58 chained WMMAs consume everything, then shfl_xor collapses
// the 16 column sums. EXEC is all-ones (32 threads, uniform control flow).
// ---------------------------------------------------------------------------
__global__ void yolo_finalize(const float* __restrict__ partials,
                              float* __restrict__ out) {
    const int lane = threadIdx.x; // 0..31
    v2f a; a.x = 1.0f; a.y = 1.0f;   // A = ones(16x4)
    v8f c = {};
    for (int k = 0; k < FIN_ITERS; ++k) {
        v2f bmat;
        bmat.x = partials[k * 64 + lane];        // B K-rows {0,2}
        bmat.y = partials[k * 64 + 32 + lane];   // B K-rows {1,3}
        c = __builtin_amdgcn_wmma_f32_16x16x4_f32(
                /*neg_a=*/false, a, /*neg_b=*/false, bmat,
                /*c_mod=*/(short)0, c, /*reuse_a=*/false, /*reuse_b=*/false);
    }
    // c[0] holds colsum[lane % 16] in every lane; xor-reduce 16 columns
    float s = c[0];
    s += __shfl_xor(s, 1, 32);
    s += __shfl_xor(s, 2, 32);
    s += __shfl_xor(s, 4, 32);
    s += __shfl_xor(s, 8, 32);
    if (lane == 0) out[0] = s * (1.0f / (float)NBATCH);
}

extern "C" void kernel_launch(void* const* d_in, const int* in_sizes, int n_in,
                              void* d_out, int out_size, void* d_ws, size_t ws_size,
                              hipStream_t stream) {
    (void)in_sizes; (void)n_in; (void)out_size; (void)ws_size;
    const float* output  = (const float*)d_in[0];
    const float* target  = (const float*)d_in[1];
    const float* anchors = (const float*)d_in[2];
    float* partials = (float*)d_ws;   // needs NPART (3712) floats = 14.5 KB
    float* loss = (float*)d_out;

    dim3 g1(P1_BLK_PER_BA, NBATCH * NANCH, 1);
    yolo_phase1<<<g1, 256, 0, stream>>>(output, target, anchors, partials);
    yolo_phase2<<<NBATCH, 64, 0, stream>>>(output, target, anchors, partials);
    yolo_finalize<<<1, 32, 0, stream>>>(partials, loss);
}